// Seq2Seq_84963043049854
// MI455X (gfx1250) — compile-verified
//
#include <hip/hip_runtime.h>
#include <cstddef>

#define BATCH 256
#define SEQX  96
#define SEQY  96
#define EDIM  256
#define HENC  512
#define HDEC  1024
#define NOUT  41

typedef float v2f __attribute__((ext_vector_type(2)));
typedef float v8f __attribute__((ext_vector_type(8)));

static __device__ __forceinline__ v8f wmma4(v2f a, v2f b, v8f c) {
  // D = A(16x4,f32) * B(4x16,f32) + C(16x16,f32)
  return __builtin_amdgcn_wmma_f32_16x16x4_f32(false, a, false, b, (short)0, c,
                                               false, false);
}

static __device__ __forceinline__ v8f splat8(float v) {
  v8f r;
#pragma unroll
  for (int i = 0; i < 8; ++i) r[i] = v;
  return r;
}

static __device__ __forceinline__ float sigm(float x) {
  return 1.0f / (1.0f + __expf(-x));
}

// Per-token input-projection tables: tbl[r][c] = emb[r] . Wih[c] + bih[c]
__global__ __launch_bounds__(256) void table_kernel(
    const float* __restrict__ emb, const float* __restrict__ Wih,
    const float* __restrict__ bih, float* __restrict__ tbl,
    int outd, int total) {
  int i = blockIdx.x * 256 + threadIdx.x;
  if (i >= total) return;
  int r = i / outd, c = i % outd;
  const float* e = emb + (size_t)r * EDIM;
  const float* w = Wih + (size_t)c * EDIM;
  float s = bih[c];
  for (int k = 0; k < EDIM; ++k) s += e[k] * w[k];
  tbl[i] = s;
}

// dec_h0 = concat(h_fwd, h_bwd)
__global__ __launch_bounds__(256) void concat_kernel(
    const float* __restrict__ henc, float* __restrict__ hdec) {
  int i = blockIdx.x * 256 + threadIdx.x;
  if (i >= BATCH * HDEC) return;
  int b = i / HDEC, c = i % HDEC;
  hdec[i] = (c < HENC) ? henc[(size_t)b * HENC + c]
                       : henc[(size_t)BATCH * HENC + (size_t)b * HENC + (c - HENC)];
}

// One GRU time step (fused 3-gate GEMM + gate math + length-mask freeze).
// Encoder mode (is_dec=0): grid covers d=0 (fwd) and d=1 (bwd, reversed gather).
// Block = 8 waves; block tile = 128 batch rows (bg) x 16 hidden cols (jt),
// each wave owns one 16x16 output tile and its r/z/n accumulators.
__global__ __launch_bounds__(256) void gru_step_kernel(
    int t, int HID, int nJT, int is_dec,
    const float* __restrict__ hcur, float* __restrict__ hnext,
    const float* __restrict__ Whh0, const float* __restrict__ Whh1,
    const float* __restrict__ bhh0, const float* __restrict__ bhh1,
    const float* __restrict__ tbl0, const float* __restrict__ tbl1,
    const int* __restrict__ toks, const int* __restrict__ lens) {
  __shared__ float As[128 * 33];  // 128 h-rows x 32 k (padded)
  __shared__ float Bs[48 * 33];   // 3 gates x 16 cols x 32 k (padded)

  const int tid = threadIdx.x;
  const int lane = tid & 31;
  const int wave = tid >> 5;

  const int bg = blockIdx.x & 1;
  const int rest = blockIdx.x >> 1;
  const int jt = rest % nJT;
  const int d = rest / nJT;
  const int j0 = jt * 16;

  const float* W = d ? Whh1 : Whh0;
  const float* bhh = d ? bhh1 : bhh0;
  const float* tbl = d ? tbl1 : tbl0;
  const float* hsrc = hcur + (size_t)d * BATCH * HID;
  float* hdst = hnext + (size_t)d * BATCH * HID;

  const int n = lane & 15;          // tile col / A row within wave tile
  const int col = j0 + n;           // hidden column
  const int kb2 = (lane >> 4) << 1; // K pair select (ISA A/B layout)

  v8f acc[3];
#pragma unroll
  for (int g = 0; g < 3; ++g) acc[g] = splat8(bhh[g * HID + col]);

  for (int k0 = 0; k0 < HID; k0 += 32) {
    for (int li = tid; li < 128 * 32; li += 256) {
      int r = li >> 5, k = li & 31;
      As[r * 33 + k] = hsrc[(size_t)(bg * 128 + r) * HID + (k0 + k)];
    }
    for (int li = tid; li < 48 * 32; li += 256) {
      int k = li & 31, idx = li >> 5;
      int g = idx >> 4, c = idx & 15;
      Bs[idx * 33 + k] = W[(size_t)(g * HID + j0 + c) * HID + (k0 + k)];
    }
    __syncthreads();

    const float* arow = &As[(wave * 16 + n) * 33];
#pragma unroll
    for (int kk = 0; kk < 32; kk += 4) {
      int kb = kk + kb2;
      v2f a;
      a.x = arow[kb];
      a.y = arow[kb + 1];
#pragma unroll
      for (int g = 0; g < 3; ++g) {
        const float* brow = &Bs[(g * 16 + n) * 33];
        v2f bf;
        bf.x = brow[kb];
        bf.y = brow[kb + 1];
        acc[g] = wmma4(a, bf, acc[g]);
      }
    }
    __syncthreads();
  }

  // Fused gate combine + mask freeze (C/D layout: n=lane&15, m=8*lane[4]+i)
#pragma unroll
  for (int i = 0; i < 8; ++i) {
    int mrow = ((lane >> 4) << 3) + i;
    int b = bg * 128 + wave * 16 + mrow;
    int L = lens[b];
    int tok;
    if (!is_dec && d == 1) {
      int tt = L - 1 - t;
      if (tt < 0) tt = 0;
      tok = toks[tt * BATCH + b];
    } else {
      tok = toks[t * BATCH + b];
    }
    const float* te = tbl + (size_t)tok * (3 * HID);
    float xr = te[col], xz = te[HID + col], xn = te[2 * HID + col];
    float hold = hsrc[(size_t)b * HID + col];
    float r = sigm(xr + acc[0][i]);
    float z = sigm(xz + acc[1][i]);
    float nn = tanhf(xn + r * acc[2][i]);
    float hnew = (1.0f - z) * nn + z * hold;
    hdst[(size_t)b * HID + col] = (t < L) ? hnew : hold;
  }
}

// Per-step output projection: logits[t] = (h @ out_W.T + out_b) * mask
__global__ __launch_bounds__(256) void out_step_kernel(
    int t, const float* __restrict__ h, const float* __restrict__ outW,
    const float* __restrict__ outb, const int* __restrict__ yls,
    float* __restrict__ out) {
  const int lane = threadIdx.x & 31;
  const int wave = threadIdx.x >> 5;
  const int wt = blockIdx.x * 8 + wave;  // 0..47 = 16 btiles x 3 ntiles
  const int bt = wt / 3, nt = wt % 3;
  const int b0 = bt * 16, n0 = nt * 16;
  const int n = lane & 15;
  const int col = n0 + n;
  const int ccol = (col < NOUT) ? col : (NOUT - 1);
  const int kb2 = (lane >> 4) << 1;

  v8f acc = splat8((col < NOUT) ? outb[col] : 0.0f);
  const float* arow = h + (size_t)(b0 + n) * HDEC;
  const float* brow = outW + (size_t)ccol * HDEC;
  for (int k0 = 0; k0 < HDEC; k0 += 4) {
    int kb = k0 + kb2;
    v2f a;
    a.x = arow[kb];
    a.y = arow[kb + 1];
    v2f bf;
    bf.x = brow[kb];
    bf.y = brow[kb + 1];
    acc = wmma4(a, bf, acc);
  }
  if (col < NOUT) {
#pragma unroll
    for (int i = 0; i < 8; ++i) {
      int b = b0 + ((lane >> 4) << 3) + i;
      float v = (t < yls[b]) ? acc[i] : 0.0f;
      out[((size_t)t * BATCH + b) * NOUT + col] = v;
    }
  }
}

extern "C" void kernel_launch(void* const* d_in, const int* in_sizes, int n_in,
                              void* d_out, int out_size, void* d_ws,
                              size_t ws_size, hipStream_t stream) {
  (void)in_sizes; (void)n_in; (void)out_size; (void)ws_size;
  const int* xs = (const int*)d_in[0];
  const int* xls = (const int*)d_in[1];
  const int* ys = (const int*)d_in[2];
  const int* yls = (const int*)d_in[3];
  const float* emb_x = (const float*)d_in[4];
  const float* emb_y = (const float*)d_in[5];
  const float* eWih_f = (const float*)d_in[6];
  const float* eWhh_f = (const float*)d_in[7];
  const float* ebih_f = (const float*)d_in[8];
  const float* ebhh_f = (const float*)d_in[9];
  const float* eWih_b = (const float*)d_in[10];
  const float* eWhh_b = (const float*)d_in[11];
  const float* ebih_b = (const float*)d_in[12];
  const float* ebhh_b = (const float*)d_in[13];
  const float* dWih = (const float*)d_in[14];
  const float* dWhh = (const float*)d_in[15];
  const float* dbih = (const float*)d_in[16];
  const float* dbhh = (const float*)d_in[17];
  const float* outW = (const float*)d_in[18];
  const float* outb = (const float*)d_in[19];
  float* out = (float*)d_out;

  float* ws = (float*)d_ws;
  size_t o = 0;
  float* tblF = ws + o; o += (size_t)29 * 3 * HENC;
  float* tblB = ws + o; o += (size_t)29 * 3 * HENC;
  float* tblD = ws + o; o += (size_t)42 * 3 * HDEC;
  float* eA   = ws + o; o += (size_t)2 * BATCH * HENC;
  float* eB   = ws + o; o += (size_t)2 * BATCH * HENC;
  float* dA   = ws + o; o += (size_t)BATCH * HDEC;
  float* dB   = ws + o;

  // Precompute per-token input-projection tables (kills the 77-GFLOP x-proj GEMMs)
  const int nF = 29 * 3 * HENC;
  table_kernel<<<(nF + 255) / 256, 256, 0, stream>>>(emb_x, eWih_f, ebih_f, tblF, 3 * HENC, nF);
  table_kernel<<<(nF + 255) / 256, 256, 0, stream>>>(emb_x, eWih_b, ebih_b, tblB, 3 * HENC, nF);
  const int nD = 42 * 3 * HDEC;
  table_kernel<<<(nD + 255) / 256, 256, 0, stream>>>(emb_y, dWih, dbih, tblD, 3 * HDEC, nD);

  (void)hipMemsetAsync(eA, 0, (size_t)2 * BATCH * HENC * sizeof(float), stream);

  // Bidirectional encoder recurrence (both directions in one grid per step)
  float* cur = eA;
  float* nxt = eB;
  for (int t = 0; t < SEQX; ++t) {
    gru_step_kernel<<<128, 256, 0, stream>>>(t, HENC, HENC / 16, 0, cur, nxt,
                                             eWhh_f, eWhh_b, ebhh_f, ebhh_b,
                                             tblF, tblB, xs, xls);
    float* tmp = cur; cur = nxt; nxt = tmp;
  }

  concat_kernel<<<(BATCH * HDEC + 255) / 256, 256, 0, stream>>>(cur, dA);

  // Decoder recurrence + fused per-step masked output projection
  float* dc = dA;
  float* dn = dB;
  for (int t = 0; t < SEQY; ++t) {
    gru_step_kernel<<<128, 256, 0, stream>>>(t, HDEC, HDEC / 16, 1, dc, dn,
                                             dWhh, dWhh, dbhh, dbhh,
                                             tblD, tblD, ys, yls);
    out_step_kernel<<<6, 256, 0, stream>>>(t, dn, outW, outb, yls, out);
    float* tmp = dc; dc = dn; dn = tmp;
  }
}